// TrainableLinguisticDensityMatrix_73529840107984
// MI455X (gfx1250) — compile-verified
//
#include <hip/hip_runtime.h>
#include <hip/hip_bf16.h>

typedef __attribute__((ext_vector_type(2))) float v2f;
typedef __attribute__((ext_vector_type(4))) float v4f;
typedef __attribute__((ext_vector_type(8))) float v8f;

#define EMB   1024
#define HDIM  8
#define TN    16          // GEMM N dimension: [r(8) | im(8)]
#define EPS   1e-10f
#define H_EPS 1e-8f

// ---------------------------------------------------------------------------
// Kernel 1: U = expm(i * (A - A^T + eps*I)), 8x8 complex, one 64-thread block.
// Scaling (2^-6) + 12-term Taylor + 6 squarings, all in LDS. Cost is trivial.
// ---------------------------------------------------------------------------
__device__ __forceinline__ v2f cmul(v2f a, v2f b) {
    return v2f{a.x * b.x - a.y * b.y, a.x * b.y + a.y * b.x};
}

__global__ void expm_kernel(const float* __restrict__ A, v2f* __restrict__ U) {
    __shared__ v2f sM[64], sP[64], sR[64], sT[64];
    const int t = threadIdx.x;           // 64 threads
    const int j = t >> 3, k = t & 7;

    const float inv = 1.0f / 64.0f;      // 2^-6 scaling
    float h = A[j * 8 + k] - A[k * 8 + j] + ((j == k) ? H_EPS : 0.0f);
    v2f m = v2f{0.0f, h * inv};          // i*H / 64
    sM[t] = m;
    sP[t] = m;
    sR[t] = v2f{(j == k) ? 1.0f : 0.0f, 0.0f} + m;   // I + M
    __syncthreads();

    // Taylor: R = sum_{n=0..12} M^n / n!
    for (int term = 2; term <= 12; ++term) {
        v2f acc = v2f{0.0f, 0.0f};
        #pragma unroll
        for (int l = 0; l < 8; ++l) acc += cmul(sP[j * 8 + l], sM[l * 8 + k]);
        acc *= (1.0f / (float)term);
        __syncthreads();
        sP[t] = acc;
        sR[t] += acc;
        __syncthreads();
    }
    // Square 6 times: R <- R @ R
    for (int s = 0; s < 6; ++s) {
        v2f acc = v2f{0.0f, 0.0f};
        #pragma unroll
        for (int l = 0; l < 8; ++l) acc += cmul(sR[j * 8 + l], sR[l * 8 + k]);
        __syncthreads();
        sR[t] = acc;
        __syncthreads();
    }
    U[t] = sR[t];
}

// ---------------------------------------------------------------------------
// Kernel 2: psi_table[vocab,16] = emb[vocab,1024] @ [W_r | W_i] + [b_r | b_i]
// One wave per 16-row tile, v_wmma_f32_16x16x4_f32, K stepped by 4.
// W staged in 64 KB LDS (sW[k*16+n]).
// ---------------------------------------------------------------------------
__global__ void vocab_gemm_wmma(const float* __restrict__ emb,
                                const float* __restrict__ W_r,
                                const float* __restrict__ b_r,
                                const float* __restrict__ W_i,
                                const float* __restrict__ b_i,
                                float* __restrict__ psi_table,
                                int vocab, int numTiles) {
    __shared__ float sW[EMB * TN];       // 64 KB
    const int t = threadIdx.x;           // 256 threads = 8 waves
    for (int idx = t; idx < EMB * TN; idx += 256) {
        int k = idx >> 4, n = idx & 15;
        sW[idx] = (n < HDIM) ? W_r[k * HDIM + n] : W_i[k * HDIM + (n - HDIM)];
    }
    __syncthreads();

    const int wave = t >> 5;
    const int lane = t & 31;
    const int tile = blockIdx.x * 8 + wave;
    if (tile >= numTiles) return;        // wave-uniform exit

    const int half = lane >> 4;          // K sub-slot (0 -> K{0,1}, 1 -> K{2,3})
    const int mn   = lane & 15;          // M for A-layout, N for B-layout
    const int mBase = tile * 16;
    int arow = mBase + mn;
    if (arow >= vocab) arow = vocab - 1; // clamp OOB rows (masked on store)
    const float* aptr = emb + (size_t)arow * EMB + 2 * half;

    v8f c = {};
    for (int k0 = 0; k0 < EMB; k0 += 4) {
        v2f a, b;
        a.x = aptr[k0];                  // A[M=mn, K=k0+2*half]
        a.y = aptr[k0 + 1];              // A[M=mn, K=k0+2*half+1]
        const int kb = k0 + 2 * half;
        b.x = sW[kb * TN + mn];          // B[K=kb,   N=mn]
        b.y = sW[(kb + 1) * TN + mn];    // B[K=kb+1, N=mn]
        c = __builtin_amdgcn_wmma_f32_16x16x4_f32(
                /*neg_a=*/false, a, /*neg_b=*/false, b,
                /*c_mod=*/(short)0, c, /*reuse_a=*/false, /*reuse_b=*/false);
    }
    const float bias = (mn < HDIM) ? b_r[mn] : b_i[mn - HDIM];
    #pragma unroll
    for (int v = 0; v < 8; ++v) {        // C/D: VGPR v -> M = v + 8*half, N = mn
        const int m = mBase + v + 8 * half;
        if (m < vocab) psi_table[(size_t)m * TN + mn] = c[v] + bias;
    }
}

// ---------------------------------------------------------------------------
// Kernel 3: per token: psi -> normalize -> phi = U psi -> out = phi phi^dag * s
// Uses U rho U^dag = (U psi)(U psi)^dag to avoid the 8x8 complex GEMMs.
// 8 threads per token; thread j writes row j (64 B) with NT float4 stores.
// ---------------------------------------------------------------------------
__global__ void token_kernel(const int* __restrict__ word_ids,
                             const float* __restrict__ psi_table,
                             const v2f* __restrict__ Uglob,
                             float* __restrict__ out, int N) {
    __shared__ v2f sU[64];
    const int t = threadIdx.x;
    if (t < 64) sU[t] = Uglob[t];
    __syncthreads();

    const int gid   = blockIdx.x * 256 + t;
    const int token = gid >> 3;
    const int j     = gid & 7;
    if (token >= N) return;

    const int word = word_ids[token];
    const v4f* p4 = (const v4f*)(psi_table + (size_t)word * TN);
    v4f q0 = p4[0], q1 = p4[1], q2 = p4[2], q3 = p4[3];
    float r[8]  = {q0.x, q0.y, q0.z, q0.w, q1.x, q1.y, q1.z, q1.w};
    float im[8] = {q2.x, q2.y, q2.z, q2.w, q3.x, q3.y, q3.z, q3.w};

    float norm = 0.0f;
    #pragma unroll
    for (int d = 0; d < 8; ++d) norm += sqrtf(r[d] * r[d] + im[d] * im[d]);
    const float s1 = 1.0f / (norm + EPS);

    float t0 = 0.0f;
    #pragma unroll
    for (int d = 0; d < 8; ++d) {
        r[d] *= s1; im[d] *= s1;
        t0 += r[d] * r[d] + im[d] * im[d];
    }
    const float d1 = t0 + EPS;           // trace of rho0

    float pr[8], pi[8];                  // phi = U @ psi_hat
    #pragma unroll
    for (int k = 0; k < 8; ++k) {
        float ar = 0.0f, ai = 0.0f;
        #pragma unroll
        for (int d = 0; d < 8; ++d) {
            const v2f u = sU[k * 8 + d];
            ar += u.x * r[d] - u.y * im[d];
            ai += u.x * im[d] + u.y * r[d];
        }
        pr[k] = ar; pi[k] = ai;
    }
    float t2a = 0.0f;
    #pragma unroll
    for (int k = 0; k < 8; ++k) t2a += pr[k] * pr[k] + pi[k] * pi[k];
    const float t2    = t2a / d1;        // trace after U (rho already /d1)
    const float scale = 1.0f / (d1 * (t2 + EPS));

    const float prj = pr[j], pij = pi[j];
    v4f* ov = (v4f*)out + (size_t)token * 32 + j * 4;   // row j: 8 complex
    #pragma unroll
    for (int kk = 0; kk < 8; kk += 2) {
        v4f v;
        v.x = (prj * pr[kk]     + pij * pi[kk])     * scale;
        v.y = (pij * pr[kk]     - prj * pi[kk])     * scale;
        v.z = (prj * pr[kk + 1] + pij * pi[kk + 1]) * scale;
        v.w = (pij * pr[kk + 1] - prj * pi[kk + 1]) * scale;
        __builtin_nontemporal_store(v, ov + (kk >> 1));
    }
}

// ---------------------------------------------------------------------------
extern "C" void kernel_launch(void* const* d_in, const int* in_sizes, int n_in,
                              void* d_out, int out_size, void* d_ws, size_t ws_size,
                              hipStream_t stream) {
    const int*   word_ids = (const int*)d_in[0];
    const float* emb      = (const float*)d_in[1];
    const float* W_r      = (const float*)d_in[2];
    const float* b_r      = (const float*)d_in[3];
    const float* W_i      = (const float*)d_in[4];
    const float* b_i      = (const float*)d_in[5];
    const float* ctxA     = (const float*)d_in[6];

    const int N     = in_sizes[0];           // BATCH*SEQ tokens
    const int vocab = in_sizes[1] / EMB;     // 50257

    // Workspace: [0,512) U (64 x complex), [1024, ...) psi_table [vocab,16] f32
    v2f*   U         = (v2f*)d_ws;
    float* psi_table = (float*)((char*)d_ws + 1024);

    expm_kernel<<<1, 64, 0, stream>>>(ctxA, U);

    const int numTiles = (vocab + 15) / 16;
    const int gBlocks  = (numTiles + 7) / 8;
    vocab_gemm_wmma<<<gBlocks, 256, 0, stream>>>(emb, W_r, b_r, W_i, b_i,
                                                 psi_table, vocab, numTiles);

    const int tBlocks = (N * 8 + 255) / 256;
    token_kernel<<<tBlocks, 256, 0, stream>>>(word_ids, psi_table, U,
                                              (float*)d_out, N);
}